// VectorQuantizerEMA_85212151152666
// MI455X (gfx1250) — compile-verified
//
#include <hip/hip_runtime.h>
#include <hip/hip_bf16.h>

typedef __attribute__((ext_vector_type(2))) float v2f;
typedef __attribute__((ext_vector_type(8))) float v8f;

#define B_ 32
#define D_ 64
#define HW_ 4096          // 64*64
#define K_ 512
#define N_ 131072         // B*H*W
#define GAMMA_ 0.99f
#define BETA_ 0.25f
#define EPS_ 1e-5f

// d_out offsets (floats), tuple order:
// out, loss, perplexity, encodings, ids_grid, distance, new_embedding, new_cluster_size, new_embed_avg
#define O_Q     ((size_t)0)
#define O_LOSS  ((size_t)8388608)
#define O_PERP  ((size_t)8388609)
#define O_ENC   ((size_t)8388610)
#define O_IDS   ((size_t)75497474)
#define O_DIST  ((size_t)75628546)
#define O_NEMB  ((size_t)142737410)
#define O_NCS   ((size_t)142770178)
#define O_NEA   ((size_t)142770690)

// ws layout (float units)
#define WS_ESQ  0         // 512 floats: sum_d emb[d,k]^2
#define WS_CNT  512       // 512 ints: counts
#define WS_SUMS 1024      // 64*512 floats: segment sums (D,K)
#define WS_PL   33792     // 8192 floats: per-block loss partials

__global__ __launch_bounds__(256) void vq_init(const float* __restrict__ emb,
                                               float* __restrict__ ws) {
  int i = blockIdx.x * blockDim.x + threadIdx.x;
  if (i < D_ * K_) ws[WS_SUMS + i] = 0.f;
  if (i < K_) {
    ((int*)(ws + WS_CNT))[i] = 0;
    float s = 0.f;
#pragma unroll 8
    for (int d = 0; d < D_; ++d) { float e = emb[(size_t)d * K_ + i]; s += e * e; }
    ws[WS_ESQ + i] = s;
  }
}

__global__ __launch_bounds__(256) void vq_main(const float* __restrict__ x,
                                               const float* __restrict__ emb,
                                               float* __restrict__ out,
                                               float* __restrict__ ws) {
  __shared__ float As[16][68];     // padded: stride 68 -> conflict-free
  __shared__ float xsq_s[16];
  __shared__ float wave_d[8][16];
  __shared__ int   wave_c[8][16];
  __shared__ int   ids_s[16];
  __shared__ float red_s[256];

  const int t   = threadIdx.x;
  const int n0  = blockIdx.x * 16;     // 16 rows of flat per block
  const int b   = n0 >> 12;            // n / (H*W)
  const int hw0 = n0 & (HW_ - 1);
  const float* esq   = ws + WS_ESQ;
  float*       sums  = ws + WS_SUMS;
  int*         counts = (int*)(ws + WS_CNT);

  // ---- stage A tile: flat[n0+r, d] = x[b, d, hw0+r]  (coalesced over r) ----
  for (int i = t; i < 16 * 64; i += 256) {
    int r = i & 15, d = i >> 4;
    As[r][d] = x[(size_t)(b * D_ + d) * HW_ + hw0 + r];
  }
  __syncthreads();
  if (t < 16) {
    float s = 0.f;
#pragma unroll 8
    for (int d = 0; d < D_; ++d) { float v = As[t][d]; s += v * v; }
    xsq_s[t] = s;
  }
  __syncthreads();

  const int w    = t >> 5;       // wave id 0..7 -> owns cols [64w, 64w+64)
  const int lane = t & 31;
  const int hi   = lane >> 4;    // lane half
  const int lm   = lane & 15;

  v8f acc[4] = {};               // 4 col tiles of 16, f32 16x16 accum each

  // ---- GEMM: dot = flat(16x64) @ emb(64x512) via v_wmma_f32_16x16x4_f32 ----
  for (int s = 0; s < 16; ++s) {
    const int ka = 4 * s + 2 * hi;                  // A: V0=k0/k2, V1=k1/k3
    v2f a = *(const v2f*)&As[lm][ka];
#pragma unroll
    for (int c = 0; c < 4; ++c) {
      const int col = ((w << 2) + c) * 16 + lm;     // B: same k striping, N=lane
      v2f bb;
      bb.x = emb[(size_t)ka * K_ + col];
      bb.y = emb[(size_t)(ka + 1) * K_ + col];
      acc[c] = __builtin_amdgcn_wmma_f32_16x16x4_f32(
          false, a, false, bb, (short)0, acc[c], false, false);
    }
  }

  // ---- distance + per-row argmin (tie -> lowest col, like jnp.argmin) ----
  float* dist = out + O_DIST;
#pragma unroll
  for (int j = 0; j < 8; ++j) {
    const int m = j + 8 * hi;              // C/D layout: VGPR j -> rows j / j+8
    const float xs = xsq_s[m];
    float bd = 3.4e38f; int bc = 0;
#pragma unroll
    for (int c = 0; c < 4; ++c) {
      const int col = ((w << 2) + c) * 16 + lm;
      const float dv = xs + esq[col] - 2.f * acc[c][j];
      dist[(size_t)(n0 + m) * K_ + col] = dv;
      if (dv < bd) { bd = dv; bc = col; }
    }
    // reduce across the 16 lanes of this half-wave
    for (int off = 1; off < 16; off <<= 1) {
      float od = __shfl_xor(bd, off, 32);
      int   oc = __shfl_xor(bc, off, 32);
      if (od < bd || (od == bd && oc < bc)) { bd = od; bc = oc; }
    }
    if (lm == 0) { wave_d[w][m] = bd; wave_c[w][m] = bc; }
  }
  __syncthreads();

  // ---- combine 8 waves (ascending col ranges keeps first-min semantics) ----
  if (t < 16) {
    float bd = wave_d[0][t]; int bc = wave_c[0][t];
    for (int w2 = 1; w2 < 8; ++w2) {
      float od = wave_d[w2][t]; int oc = wave_c[w2][t];
      if (od < bd || (od == bd && oc < bc)) { bd = od; bc = oc; }
    }
    ids_s[t] = bc;
    out[O_IDS + n0 + t] = (float)bc;
    atomicAdd(&counts[bc], 1);
  }
  __syncthreads();

  // ---- quantized output (NCHW), loss partial, EMA segment sums ----
  float lsum = 0.f;
  for (int i = t; i < 16 * 64; i += 256) {
    int r = i & 15, d = i >> 4;
    const int id = ids_s[r];
    const float q  = emb[(size_t)d * K_ + id];
    const float xv = As[r][d];
    out[O_Q + (size_t)(b * D_ + d) * HW_ + hw0 + r] = q;  // straight-through == q
    const float df = xv - q;
    lsum += df * df;
    atomicAdd(&sums[(size_t)d * K_ + id], xv);
  }
  red_s[t] = lsum;
  __syncthreads();
  for (int off = 128; off > 0; off >>= 1) {
    if (t < off) red_s[t] += red_s[t + off];
    __syncthreads();
  }
  if (t == 0) ws[WS_PL + blockIdx.x] = red_s[0];

  // ---- one-hot encodings (coalesced over k) ----
  for (int i = t; i < 16 * K_; i += 256) {
    int r = i >> 9, k = i & (K_ - 1);
    out[O_ENC + (size_t)(n0 + r) * K_ + k] = (k == ids_s[r]) ? 1.f : 0.f;
  }
}

__global__ __launch_bounds__(256) void vq_loss(float* __restrict__ out,
                                               const float* __restrict__ ws) {
  __shared__ float r[256];
  int t = threadIdx.x;
  float s = 0.f;
  for (int i = t; i < 8192; i += 256) s += ws[WS_PL + i];
  r[t] = s;
  __syncthreads();
  for (int off = 128; off > 0; off >>= 1) {
    if (t < off) r[t] += r[t + off];
    __syncthreads();
  }
  if (t == 0) out[O_LOSS] = BETA_ * r[0] / 8388608.f;  // mean over B*H*W*D
}

__global__ __launch_bounds__(512) void vq_finalize(const float* __restrict__ cs,
                                                   const float* __restrict__ ea,
                                                   float* __restrict__ out,
                                                   const float* __restrict__ ws) {
  __shared__ float ncs_s[512];
  __shared__ float term_s[512];
  const int t = threadIdx.x;
  const int* counts = (const int*)(ws + WS_CNT);
  const float* sums = ws + WS_SUMS;

  const float c   = (float)counts[t];
  const float ncs = cs[t] * GAMMA_ + (1.f - GAMMA_) * c;
  out[O_NCS + t] = ncs;
  const float p = c / (float)N_;
  ncs_s[t]  = ncs;
  term_s[t] = p * logf(p + 1e-10f);
  __syncthreads();
  for (int off = 256; off > 0; off >>= 1) {
    if (t < off) { ncs_s[t] += ncs_s[t + off]; term_s[t] += term_s[t + off]; }
    __syncthreads();
  }
  const float n = ncs_s[0];
  if (t == 0) out[O_PERP] = expf(-term_s[0]);

  const float csn = (ncs + EPS_) / (n + (float)K_ * EPS_) * n;
  for (int d = 0; d < D_; ++d) {
    const size_t idx = (size_t)d * K_ + t;
    const float nea = ea[idx] * GAMMA_ + (1.f - GAMMA_) * sums[idx];
    out[O_NEA + idx]  = nea;
    out[O_NEMB + idx] = nea / csn;
  }
}

extern "C" void kernel_launch(void* const* d_in, const int* in_sizes, int n_in,
                              void* d_out, int out_size, void* d_ws, size_t ws_size,
                              hipStream_t stream) {
  const float* x   = (const float*)d_in[0];   // [B, D, H, W]
  const float* emb = (const float*)d_in[1];   // [D, K]
  const float* cs  = (const float*)d_in[2];   // [K]
  const float* ea  = (const float*)d_in[3];   // [D, K]
  float* out = (float*)d_out;
  float* ws  = (float*)d_ws;

  vq_init<<<128, 256, 0, stream>>>(emb, ws);                 // 32768 threads
  vq_main<<<N_ / 16, 256, 0, stream>>>(x, emb, out, ws);     // 8192 blocks
  vq_loss<<<1, 256, 0, stream>>>(out, ws);
  vq_finalize<<<1, 512, 0, stream>>>(cs, ea, out, ws);
}